// CAM_Module_27264452395644
// MI455X (gfx1250) — compile-verified
//
#include <hip/hip_runtime.h>
#include <stdint.h>

// CAM module (channel attention), x:(16,512,64,64) fp32.
//  q_b = per-batch x buffer viewed as N x C row-major (flat reshape, no copy)
//  E_b = Q^T Q (fp32 WMMA 16x16x4), mx = colmax of batch-0 E,
//  att = softmax_d(mx[d]-E), out = att @ Q^T, y = gamma*out + x (fused).
// GEMM staging uses the Tensor Data Mover (tensor_load_to_lds) with padded-row
// tiles, double-buffered on TENSORcnt; fragments come from LDS as b64 pairs.

typedef float v2f __attribute__((ext_vector_type(2)));
typedef float v8f __attribute__((ext_vector_type(8)));
typedef unsigned int u32x4 __attribute__((ext_vector_type(4)));
typedef int i32x4 __attribute__((ext_vector_type(4)));
typedef int i32x8 __attribute__((ext_vector_type(8)));

#define B_ 16
#define C_ 512
#define N_ 4096
#define KT 32
#define LP1 132   // kernel1 LDS row: 128 + 4 floats (TDM pad_interval=6, pad_amount=3)
#define LP4 34    // kernel4 LDS row: 32 + 2 floats  (TDM pad_interval=4, pad_amount=1)

#if defined(__has_builtin)
#if __has_builtin(__builtin_amdgcn_tensor_load_to_lds)
#define HAVE_TDM 1
#endif
#endif
#ifndef HAVE_TDM
#define HAVE_TDM 0
#endif

__device__ __forceinline__ unsigned lds_off(const void* p) {
  // generic pointers to LDS carry the byte offset in the low 32 bits
  return (unsigned)(unsigned long long)(size_t)p;
}

#if HAVE_TDM
// 2-D tile DMA: global (tile_h rows x tile_w elems, row stride `stride` elems,
// fp32) -> LDS at lds_addr, with pad_amount+1 DWORDs inserted after every
// 2^(pad_interval+1) DWORDs (ISA 8.4). Completion tracked by TENSORcnt.
__device__ __forceinline__ void tdm_load_2d(unsigned lds_addr, const float* gtile,
                                            unsigned tile_w, unsigned tile_h,
                                            unsigned stride,
                                            unsigned tensor_w, unsigned tensor_h,
                                            unsigned pad_interval, unsigned pad_amount) {
  unsigned long long ga = (unsigned long long)(size_t)gtile;
  u32x4 g0;
  g0[0] = 1u;                                                  // count=1, user desc
  g0[1] = lds_addr;                                            // lds_addr
  g0[2] = (unsigned)ga;                                        // global_addr[31:0]
  g0[3] = (unsigned)((ga >> 32) & 0x01FFFFFFu) | 0x80000000u;  // addr[56:32] | type=2
  i32x8 g1;
  g1[0] = (int)((2u << 16)                  // data_size = 4 bytes
              | (1u << 20)                  // pad_enable
              | (pad_interval << 22)
              | (pad_amount << 25));
  g1[1] = (int)(tensor_w << 16);                               // tensor_dim0[15:0]
  g1[2] = (int)((tensor_w >> 16) | (tensor_h << 16));          // dim0 hi | dim1 lo
  g1[3] = (int)((tensor_h >> 16) | (tile_w << 16));            // dim1 hi | tile_dim0
  g1[4] = (int)(tile_h & 0xFFFFu);                             // tile_dim1 (tile_dim2=0)
  g1[5] = (int)stride;                                         // tensor_dim0_stride lo
  g1[6] = 0;
  g1[7] = 0;
  i32x4 z4 = {0, 0, 0, 0};
#if __clang_major__ >= 23
  i32x8 z8 = {0, 0, 0, 0, 0, 0, 0, 0};
  __builtin_amdgcn_tensor_load_to_lds(g0, g1, z4, z4, z8, 0);
#else
  __builtin_amdgcn_tensor_load_to_lds(g0, g1, z4, z4, 0);
#endif
}
#endif  // HAVE_TDM

// ---------------------------------------------------------------------------
// Kernel 1: E_b = Q_b^T Q_b (512x512, K=4096). WG = 128 thr = 4 waves (2x2),
// WG tile 128x128, each wave 64x64 = 4x4 WMMA accumulators.
// LDS tiles [k][132] (exact TDM padded layout), double-buffered via TDM.
// ---------------------------------------------------------------------------
__global__ __launch_bounds__(128) void cam_energy_kernel(const float* __restrict__ x,
                                                         float* __restrict__ E) {
  const int b   = blockIdx.z;
  const int cm0 = blockIdx.y * 128;
  const int cn0 = blockIdx.x * 128;
  const float* __restrict__ Q  = x + (size_t)b * C_ * N_;   // N_ x C_ row-major
  float* __restrict__       Eb = E + (size_t)b * C_ * C_;

  __shared__ float As[2][KT][LP1];
  __shared__ float Bs[2][KT][LP1];

  const int t = threadIdx.x, lane = t & 31, wave = t >> 5;
  const int wm = wave & 1, wn = wave >> 1;
  const int half = lane >> 4, l16 = lane & 15;

  const v8f vzero = {0.f, 0.f, 0.f, 0.f, 0.f, 0.f, 0.f, 0.f};
  v8f acc[4][4];
#pragma unroll
  for (int mi = 0; mi < 4; ++mi)
#pragma unroll
    for (int ni = 0; ni < 4; ++ni) acc[mi][ni] = vzero;

  const int S = N_ / KT;   // 128 K-stages

#if HAVE_TDM
  if (wave == 0) {   // uniform branch: EXEC stays full inside each wave
    tdm_load_2d(lds_off(&As[0][0][0]), Q + cm0, 128, KT, C_, C_, N_, 6, 3);
    tdm_load_2d(lds_off(&Bs[0][0][0]), Q + cn0, 128, KT, C_, C_, N_, 6, 3);
  }
#endif

  for (int s = 0; s < S; ++s) {
#if HAVE_TDM
    if (wave == 0) {
      if (s + 1 < S) {   // prefetch next stage, then wait for current pair
        const float* rowp = Q + (size_t)(s + 1) * KT * C_;
        tdm_load_2d(lds_off(&As[(s + 1) & 1][0][0]), rowp + cm0, 128, KT, C_, C_, N_, 6, 3);
        tdm_load_2d(lds_off(&Bs[(s + 1) & 1][0][0]), rowp + cn0, 128, KT, C_, C_, N_, 6, 3);
        __builtin_amdgcn_s_wait_tensorcnt(2);
      } else {
        __builtin_amdgcn_s_wait_tensorcnt(0);
      }
    }
#else
    {
      const int k0 = s * KT;
#pragma unroll
      for (int i = 0; i < 8; ++i) {
        int v  = t + i * 128;     // 1024 float4 slots per tile
        int r  = v >> 5;
        int cv = (v & 31) << 2;
        const float* rowp = Q + (size_t)(k0 + r) * C_;
        *reinterpret_cast<float4*>(&As[s & 1][r][cv]) =
            *reinterpret_cast<const float4*>(rowp + cm0 + cv);
        *reinterpret_cast<float4*>(&Bs[s & 1][r][cv]) =
            *reinterpret_cast<const float4*>(rowp + cn0 + cv);
      }
    }
#endif
    __syncthreads();

    const float(*A)[LP1]  = As[s & 1];
    const float(*Bt)[LP1] = Bs[s & 1];
#pragma unroll
    for (int kk = 0; kk < KT; kk += 4) {
      const int ka = kk + half * 2;   // lane half selects K pair (ISA A/B layout)
      v2f af[4], bf[4];
#pragma unroll
      for (int mi = 0; mi < 4; ++mi) {
        int m = wm * 64 + mi * 16 + l16;
        af[mi][0] = A[ka][m];
        af[mi][1] = A[ka + 1][m];
      }
#pragma unroll
      for (int ni = 0; ni < 4; ++ni) {
        int n = wn * 64 + ni * 16 + l16;
        bf[ni][0] = Bt[ka][n];
        bf[ni][1] = Bt[ka + 1][n];
      }
#pragma unroll
      for (int mi = 0; mi < 4; ++mi)
#pragma unroll
        for (int ni = 0; ni < 4; ++ni)
          acc[mi][ni] = __builtin_amdgcn_wmma_f32_16x16x4_f32(
              false, af[mi], false, bf[ni], (short)0, acc[mi][ni], false, false);
    }
    __syncthreads();
  }

  // C/D layout: VGPR r = row (r + 8*lane_half), col = lane%16
#pragma unroll
  for (int mi = 0; mi < 4; ++mi)
#pragma unroll
    for (int ni = 0; ni < 4; ++ni) {
      int col     = cn0 + wn * 64 + ni * 16 + l16;
      int rowbase = cm0 + wm * 64 + mi * 16 + half * 8;
#pragma unroll
      for (int r = 0; r < 8; ++r)
        Eb[(size_t)(rowbase + r) * C_ + col] = acc[mi][ni][r];
    }
}

// ---------------------------------------------------------------------------
// Kernel 2: mx[d] = max_c E[0][c][d] (batch 0 only, per reference's [0])
// ---------------------------------------------------------------------------
__global__ __launch_bounds__(256) void cam_colmax_kernel(const float* __restrict__ E0,
                                                         float* __restrict__ mx) {
  int d = blockIdx.x * blockDim.x + threadIdx.x;
  float m = -3.402823466e+38f;
  for (int c = 0; c < C_; ++c)
    m = fmaxf(m, E0[(size_t)c * C_ + d]);
  mx[d] = m;
}

// ---------------------------------------------------------------------------
// Kernel 3: in-place row softmax of (mx[d] - E[b][c][d]) over d.
// One wave32 per row; 16 elems/lane; shfl_xor reductions.
// ---------------------------------------------------------------------------
__global__ __launch_bounds__(256) void cam_softmax_kernel(float* __restrict__ E,
                                                          const float* __restrict__ mx) {
  const int lane = threadIdx.x & 31;
  const int wave = threadIdx.x >> 5;
  float* __restrict__ e = E + ((size_t)blockIdx.x * 8 + wave) * C_;

  float v[16];
  float m = -3.402823466e+38f;
#pragma unroll
  for (int i = 0; i < 16; ++i) {
    int d = lane + i * 32;
    v[i]  = mx[d] - e[d];
    m     = fmaxf(m, v[i]);
  }
#pragma unroll
  for (int off = 16; off > 0; off >>= 1)
    m = fmaxf(m, __shfl_xor(m, off, 32));
  float s = 0.f;
#pragma unroll
  for (int i = 0; i < 16; ++i) { v[i] = __expf(v[i] - m); s += v[i]; }
#pragma unroll
  for (int off = 16; off > 0; off >>= 1)
    s += __shfl_xor(s, off, 32);
  const float inv = 1.0f / s;
#pragma unroll
  for (int i = 0; i < 16; ++i)
    e[lane + i * 32] = v[i] * inv;
}

// ---------------------------------------------------------------------------
// Kernel 4: out_b = Att_b (512x512) @ Q_b^T (512x4096); y = gamma*out + x.
// Both operands' global rows run along K -> TDM row-major tiles [row][34] put
// the K-pair contiguous: every fragment is one conflict-free ds_load_b64.
// ---------------------------------------------------------------------------
__global__ __launch_bounds__(128) void cam_out_kernel(const float* __restrict__ x,
                                                      const float* __restrict__ Att,
                                                      const float* __restrict__ gamma,
                                                      float* __restrict__ y) {
  const int b   = blockIdx.z;
  const int cm0 = blockIdx.y * 128;   // output channel base
  const int n0  = blockIdx.x * 128;   // output spatial base
  const float* __restrict__ Ab = Att + (size_t)b * C_ * C_;
  const float* __restrict__ Qb = x + (size_t)b * C_ * N_;

  __shared__ float As[2][128][LP4];   // As[.][m][k] = Att[cm0+m][k0+k]
  __shared__ float Bs[2][128][LP4];   // Bs[.][n][k] = Q[n0+n][k0+k] = Q^T[k][n]

  const int t = threadIdx.x, lane = t & 31, wave = t >> 5;
  const int wm = wave & 1, wn = wave >> 1;
  const int half = lane >> 4, l16 = lane & 15;

  const v8f vzero = {0.f, 0.f, 0.f, 0.f, 0.f, 0.f, 0.f, 0.f};
  v8f acc[4][4];
#pragma unroll
  for (int mi = 0; mi < 4; ++mi)
#pragma unroll
    for (int ni = 0; ni < 4; ++ni) acc[mi][ni] = vzero;

  const int S = C_ / KT;   // 16 K-stages

#if HAVE_TDM
  if (wave == 0) {
    tdm_load_2d(lds_off(&As[0][0][0]), Ab + (size_t)cm0 * C_, KT, 128, C_, C_, C_, 4, 1);
    tdm_load_2d(lds_off(&Bs[0][0][0]), Qb + (size_t)n0 * C_, KT, 128, C_, C_, N_, 4, 1);
  }
#endif

  for (int s = 0; s < S; ++s) {
#if HAVE_TDM
    if (wave == 0) {
      if (s + 1 < S) {
        const int k0n = (s + 1) * KT;
        tdm_load_2d(lds_off(&As[(s + 1) & 1][0][0]), Ab + (size_t)cm0 * C_ + k0n,
                    KT, 128, C_, C_, C_, 4, 1);
        tdm_load_2d(lds_off(&Bs[(s + 1) & 1][0][0]), Qb + (size_t)n0 * C_ + k0n,
                    KT, 128, C_, C_, N_, 4, 1);
        __builtin_amdgcn_s_wait_tensorcnt(2);
      } else {
        __builtin_amdgcn_s_wait_tensorcnt(0);
      }
    }
#else
    {
      const int k0 = s * KT;
#pragma unroll
      for (int i = 0; i < 8; ++i) {
        int v  = t + i * 128;     // 1024 float4 slots per tile
        int m  = v >> 3;          // row 0..127
        int kv = (v & 7) << 2;    // k offset 0..28
        float4 fa = *reinterpret_cast<const float4*>(Ab + (size_t)(cm0 + m) * C_ + k0 + kv);
        *reinterpret_cast<float2*>(&As[s & 1][m][kv])     = make_float2(fa.x, fa.y);
        *reinterpret_cast<float2*>(&As[s & 1][m][kv + 2]) = make_float2(fa.z, fa.w);
        float4 fb = *reinterpret_cast<const float4*>(Qb + (size_t)(n0 + m) * C_ + k0 + kv);
        *reinterpret_cast<float2*>(&Bs[s & 1][m][kv])     = make_float2(fb.x, fb.y);
        *reinterpret_cast<float2*>(&Bs[s & 1][m][kv + 2]) = make_float2(fb.z, fb.w);
      }
    }
#endif
    __syncthreads();

    const float(*A)[LP4]  = As[s & 1];
    const float(*Bt)[LP4] = Bs[s & 1];
#pragma unroll
    for (int kk = 0; kk < KT; kk += 4) {
      const int ka = kk + half * 2;
      v2f af[4], bf[4];
#pragma unroll
      for (int mi = 0; mi < 4; ++mi) {
        int m = wm * 64 + mi * 16 + l16;
        af[mi] = *reinterpret_cast<const v2f*>(&A[m][ka]);   // one ds_load_b64
      }
#pragma unroll
      for (int ni = 0; ni < 4; ++ni) {
        int n = wn * 64 + ni * 16 + l16;
        bf[ni] = *reinterpret_cast<const v2f*>(&Bt[n][ka]);  // one ds_load_b64
      }
#pragma unroll
      for (int mi = 0; mi < 4; ++mi)
#pragma unroll
        for (int ni = 0; ni < 4; ++ni)
          acc[mi][ni] = __builtin_amdgcn_wmma_f32_16x16x4_f32(
              false, af[mi], false, bf[ni], (short)0, acc[mi][ni], false, false);
    }
    __syncthreads();
  }

  const float g = *gamma;
#pragma unroll
  for (int mi = 0; mi < 4; ++mi)
#pragma unroll
    for (int ni = 0; ni < 4; ++ni) {
      int n = n0 + wn * 64 + ni * 16 + l16;
      int c = cm0 + wm * 64 + mi * 16 + half * 8;
#pragma unroll
      for (int r = 0; r < 8; ++r) {
        size_t idx = (size_t)b * C_ * N_ + (size_t)(c + r) * N_ + n;
        y[idx] = g * acc[mi][ni][r] + x[idx];   // fused gamma*out + x
      }
    }
}

// ---------------------------------------------------------------------------
extern "C" void kernel_launch(void* const* d_in, const int* in_sizes, int n_in,
                              void* d_out, int out_size, void* d_ws, size_t ws_size,
                              hipStream_t stream) {
  const float* x     = (const float*)d_in[0];   // (16,512,64,64) fp32
  const float* gamma = (const float*)d_in[1];   // scalar fp32
  float*       y     = (float*)d_out;

  float* E  = (float*)d_ws;                     // B*C*C floats = 16.78 MB
  float* mx = E + (size_t)B_ * C_ * C_;         // 512 floats

  cam_energy_kernel<<<dim3(C_ / 128, C_ / 128, B_), 128, 0, stream>>>(x, E);
  cam_colmax_kernel<<<dim3(C_ / 256), 256, 0, stream>>>(E, mx);
  cam_softmax_kernel<<<dim3(B_ * C_ / 8), 256, 0, stream>>>(E, mx);
  cam_out_kernel<<<dim3(N_ / 128, C_ / 128, B_), 128, 0, stream>>>(x, E, gamma, y);
}